// SO2_Linear_24927990186029
// MI455X (gfx1250) — compile-verified
//
#include <hip/hip_runtime.h>

typedef __attribute__((ext_vector_type(2))) float v2f;
typedef __attribute__((ext_vector_type(8))) float v8f;

#define NBLK     16      // samples per workgroup
#define NTHREADS 256     // 8 waves (wave32)

// LDS layout (float offsets)
#define OFF_WIG 0        // [16][256]  per-sample 16x16 wigner
#define OFF_X0  4096     // [16][256]  m=0 block   (X, then Y in-place)
#define OFF_X1  8192     // [32][192]  m=1 block (rows 0-15: -m, 16-31: +m)
#define OFF_X2  14336    // [32][128]  m=2 block
#define OFF_X3  18432    // [32][64]   m=3 block
#define LDS_FLOATS 20480 // 80 KB

__global__ __launch_bounds__(NTHREADS)
void so2_linear_kernel(const float* __restrict__ x,
                       const float* __restrict__ wigner,
                       const float* __restrict__ W0,
                       const float* __restrict__ b0,
                       const float* __restrict__ W1,
                       const float* __restrict__ W2,
                       const float* __restrict__ W3,
                       float* __restrict__ out,
                       int N)
{
    __shared__ float ls[LDS_FLOATS];
    const int tid = threadIdx.x;
    const int n0  = blockIdx.x * NBLK;

    // ---------------- stage wigner blocks into LDS ----------------
    #pragma unroll
    for (int j = 0; j < 16; ++j) {
        int q = tid + NTHREADS * j;          // 0..4095
        int s = q >> 8;
        int n = n0 + s; if (n >= N) n = N - 1;
        ls[OFF_WIG + q] = wigner[(size_t)n * 256 + (q & 255)];
    }
    __syncthreads();

    // ---------------- phase 1: forward rotation -> X matrices ----------------
    #pragma unroll
    for (int j = 0; j < 4; ++j) {
        int p   = tid + NTHREADS * j;        // 0..1023
        int s   = p >> 6;
        int mul = p & 63;
        int n   = n0 + s; if (n >= N) n = N - 1;
        const float* xr = x + (size_t)n * 1024;
        const float* wg = &ls[OFF_WIG + s * 256];

        // l=0 (no rotation)
        ls[OFF_X0 + s * 256 + mul] = xr[mul];

        // l=1: 64x3 @ D1(3x3)
        {
            float rt[3] = {0.f, 0.f, 0.f};
            #pragma unroll
            for (int c = 0; c < 3; ++c) {
                float xv = xr[64 + mul * 3 + c];
                const float* wr = wg + (1 + c) * 16 + 1;
                #pragma unroll
                for (int d = 0; d < 3; ++d) rt[d] = fmaf(xv, wr[d], rt[d]);
            }
            ls[OFF_X0 + s * 256 + 64 + mul]   = rt[1];
            ls[OFF_X1 + s * 192 + mul]        = rt[0];
            ls[OFF_X1 + (16 + s) * 192 + mul] = rt[2];
        }
        // l=2: 64x5 @ D2(5x5)
        {
            float rt[5] = {0.f, 0.f, 0.f, 0.f, 0.f};
            #pragma unroll
            for (int c = 0; c < 5; ++c) {
                float xv = xr[256 + mul * 5 + c];
                const float* wr = wg + (4 + c) * 16 + 4;
                #pragma unroll
                for (int d = 0; d < 5; ++d) rt[d] = fmaf(xv, wr[d], rt[d]);
            }
            ls[OFF_X0 + s * 256 + 128 + mul]       = rt[2];
            ls[OFF_X1 + s * 192 + 64 + mul]        = rt[1];
            ls[OFF_X1 + (16 + s) * 192 + 64 + mul] = rt[3];
            ls[OFF_X2 + s * 128 + mul]             = rt[0];
            ls[OFF_X2 + (16 + s) * 128 + mul]      = rt[4];
        }
        // l=3: 64x7 @ D3(7x7)
        {
            float rt[7] = {0.f, 0.f, 0.f, 0.f, 0.f, 0.f, 0.f};
            #pragma unroll
            for (int c = 0; c < 7; ++c) {
                float xv = xr[576 + mul * 7 + c];
                const float* wr = wg + (9 + c) * 16 + 9;
                #pragma unroll
                for (int d = 0; d < 7; ++d) rt[d] = fmaf(xv, wr[d], rt[d]);
            }
            ls[OFF_X0 + s * 256 + 192 + mul]        = rt[3];
            ls[OFF_X1 + s * 192 + 128 + mul]        = rt[2];
            ls[OFF_X1 + (16 + s) * 192 + 128 + mul] = rt[4];
            ls[OFF_X2 + s * 128 + 64 + mul]         = rt[1];
            ls[OFF_X2 + (16 + s) * 128 + 64 + mul]  = rt[5];
            ls[OFF_X3 + s * 64 + mul]               = rt[0];
            ls[OFF_X3 + (16 + s) * 64 + mul]        = rt[6];
        }
    }
    __syncthreads();

    // ---------------- phase 2: SO(2) weight GEMMs via f32 WMMA ----------------
    // Per-block tile distribution over 8 waves (all counts divide evenly):
    //   m0: 16 tiles -> 2/wave   m1: 24 -> 3/wave   m2: 16 -> 2/wave   m3: 8 -> 1/wave
    // K is a compile-time constant per block; W pointers used directly so the
    // loads stay on the GLOBAL path and the K-loops unroll by 8 (m3 fully)/pipeline.
    const int wid  = __builtin_amdgcn_readfirstlane(tid >> 5);  // uniform wave id
    const int lane = tid & 31;
    const int hi   = lane >> 4;        // 0: K pair {k,k+1}, 1: {k+2,k+3}
    const int ln   = lane & 15;        // A row / B col within tile
    const int koff = 2 * hi;

    v8f acc0[2], acc1[3], acc2[2], acc3[1];

    // ---- m0: M=16, N=256, K=256
    #pragma unroll
    for (int ti = 0; ti < 2; ++ti) {
        const int col0 = (wid + ti * 8) << 4;
        const float* Xp = &ls[OFF_X0 + ln * 256 + koff];
        const float* Wp = W0 + (size_t)(col0 + ln) * 256 + koff;
        v8f c = {0.f, 0.f, 0.f, 0.f, 0.f, 0.f, 0.f, 0.f};
        #pragma unroll 8
        for (int k = 0; k < 256; k += 4) {
            v2f a  = *(const v2f*)(Xp + k);
            v2f bv = *(const v2f*)(Wp + k);
            c = __builtin_amdgcn_wmma_f32_16x16x4_f32(false, a, false, bv,
                                                      (short)0, c, false, false);
        }
        acc0[ti] = c;
    }
    // ---- m1: M=32 (rows 0/16), N=192, K=192
    #pragma unroll
    for (int ti = 0; ti < 3; ++ti) {
        const int t    = wid + ti * 8;            // 0..23
        const int row0 = (t >= 12) ? 16 : 0;
        const int col0 = (t % 12) << 4;
        const float* Xp = &ls[OFF_X1 + (row0 + ln) * 192 + koff];
        const float* Wp = W1 + (size_t)(col0 + ln) * 192 + koff;
        v8f c = {0.f, 0.f, 0.f, 0.f, 0.f, 0.f, 0.f, 0.f};
        #pragma unroll 8
        for (int k = 0; k < 192; k += 4) {
            v2f a  = *(const v2f*)(Xp + k);
            v2f bv = *(const v2f*)(Wp + k);
            c = __builtin_amdgcn_wmma_f32_16x16x4_f32(false, a, false, bv,
                                                      (short)0, c, false, false);
        }
        acc1[ti] = c;
    }
    // ---- m2: M=32, N=128, K=128
    #pragma unroll
    for (int ti = 0; ti < 2; ++ti) {
        const int t    = wid + ti * 8;            // 0..15
        const int row0 = (t >= 8) ? 16 : 0;
        const int col0 = (t & 7) << 4;
        const float* Xp = &ls[OFF_X2 + (row0 + ln) * 128 + koff];
        const float* Wp = W2 + (size_t)(col0 + ln) * 128 + koff;
        v8f c = {0.f, 0.f, 0.f, 0.f, 0.f, 0.f, 0.f, 0.f};
        #pragma unroll 8
        for (int k = 0; k < 128; k += 4) {
            v2f a  = *(const v2f*)(Xp + k);
            v2f bv = *(const v2f*)(Wp + k);
            c = __builtin_amdgcn_wmma_f32_16x16x4_f32(false, a, false, bv,
                                                      (short)0, c, false, false);
        }
        acc2[ti] = c;
    }
    // ---- m3: M=32, N=64, K=64
    {
        const int t    = wid;                     // 0..7
        const int row0 = (t >= 4) ? 16 : 0;
        const int col0 = (t & 3) << 4;
        const float* Xp = &ls[OFF_X3 + (row0 + ln) * 64 + koff];
        const float* Wp = W3 + (size_t)(col0 + ln) * 64 + koff;
        v8f c = {0.f, 0.f, 0.f, 0.f, 0.f, 0.f, 0.f, 0.f};
        #pragma unroll
        for (int k = 0; k < 64; k += 4) {
            v2f a  = *(const v2f*)(Xp + k);
            v2f bv = *(const v2f*)(Wp + k);
            c = __builtin_amdgcn_wmma_f32_16x16x4_f32(false, a, false, bv,
                                                      (short)0, c, false, false);
        }
        acc3[0] = c;
    }
    __syncthreads();

    // ---- write Y tiles back in-place over X (shapes match; W square), fuse m0 bias
    #pragma unroll
    for (int ti = 0; ti < 2; ++ti) {
        const int col0 = (wid + ti * 8) << 4;
        float bias = b0[col0 + ln];
        #pragma unroll
        for (int r = 0; r < 8; ++r)
            ls[OFF_X0 + (r + 8 * hi) * 256 + col0 + ln] = acc0[ti][r] + bias;
    }
    #pragma unroll
    for (int ti = 0; ti < 3; ++ti) {
        const int t    = wid + ti * 8;
        const int row0 = (t >= 12) ? 16 : 0;
        const int col0 = (t % 12) << 4;
        #pragma unroll
        for (int r = 0; r < 8; ++r)
            ls[OFF_X1 + (row0 + r + 8 * hi) * 192 + col0 + ln] = acc1[ti][r];
    }
    #pragma unroll
    for (int ti = 0; ti < 2; ++ti) {
        const int t    = wid + ti * 8;
        const int row0 = (t >= 8) ? 16 : 0;
        const int col0 = (t & 7) << 4;
        #pragma unroll
        for (int r = 0; r < 8; ++r)
            ls[OFF_X2 + (row0 + r + 8 * hi) * 128 + col0 + ln] = acc2[ti][r];
    }
    {
        const int row0 = (wid >= 4) ? 16 : 0;
        const int col0 = (wid & 3) << 4;
        #pragma unroll
        for (int r = 0; r < 8; ++r)
            ls[OFF_X3 + (row0 + r + 8 * hi) * 64 + col0 + ln] = acc3[0][r];
    }
    __syncthreads();

    // ---------------- phase 4: inverse rotation + store ----------------
    #pragma unroll
    for (int j = 0; j < 4; ++j) {
        int p   = tid + NTHREADS * j;
        int s   = p >> 6;
        int mul = p & 63;
        int n   = n0 + s;
        if (n >= N) continue;
        const float* wg = &ls[OFF_WIG + s * 256];
        const float* y0 = &ls[OFF_X0 + s * 256];
        float* op = out + (size_t)n * 1024;

        // l=0
        op[mul] = y0[mul];
        // l=1: comps @ D1^T
        {
            float c0 = ls[OFF_X1 + s * 192 + mul];
            float c1 = y0[64 + mul];
            float c2 = ls[OFF_X1 + (16 + s) * 192 + mul];
            #pragma unroll
            for (int d = 0; d < 3; ++d) {
                const float* wr = wg + (1 + d) * 16 + 1;
                op[64 + mul * 3 + d] = c0 * wr[0] + c1 * wr[1] + c2 * wr[2];
            }
        }
        // l=2
        {
            float c0 = ls[OFF_X2 + s * 128 + mul];
            float c1 = ls[OFF_X1 + s * 192 + 64 + mul];
            float c2 = y0[128 + mul];
            float c3 = ls[OFF_X1 + (16 + s) * 192 + 64 + mul];
            float c4 = ls[OFF_X2 + (16 + s) * 128 + mul];
            #pragma unroll
            for (int d = 0; d < 5; ++d) {
                const float* wr = wg + (4 + d) * 16 + 4;
                op[256 + mul * 5 + d] =
                    c0 * wr[0] + c1 * wr[1] + c2 * wr[2] + c3 * wr[3] + c4 * wr[4];
            }
        }
        // l=3
        {
            float c0 = ls[OFF_X3 + s * 64 + mul];
            float c1 = ls[OFF_X2 + s * 128 + 64 + mul];
            float c2 = ls[OFF_X1 + s * 192 + 128 + mul];
            float c3 = y0[192 + mul];
            float c4 = ls[OFF_X1 + (16 + s) * 192 + 128 + mul];
            float c5 = ls[OFF_X2 + (16 + s) * 128 + 64 + mul];
            float c6 = ls[OFF_X3 + (16 + s) * 64 + mul];
            #pragma unroll
            for (int d = 0; d < 7; ++d) {
                const float* wr = wg + (9 + d) * 16 + 9;
                op[576 + mul * 7 + d] =
                    c0 * wr[0] + c1 * wr[1] + c2 * wr[2] + c3 * wr[3] +
                    c4 * wr[4] + c5 * wr[5] + c6 * wr[6];
            }
        }
    }
}

extern "C" void kernel_launch(void* const* d_in, const int* in_sizes, int n_in,
                              void* d_out, int out_size, void* d_ws, size_t ws_size,
                              hipStream_t stream) {
    (void)n_in; (void)out_size; (void)d_ws; (void)ws_size;
    const float* x   = (const float*)d_in[0];
    // d_in[1] = R, unused by the reference
    const float* wig = (const float*)d_in[2];
    const float* W0  = (const float*)d_in[3];
    const float* b0  = (const float*)d_in[4];
    const float* W1  = (const float*)d_in[5];
    const float* W2  = (const float*)d_in[6];
    const float* W3  = (const float*)d_in[7];
    float* out = (float*)d_out;

    const int N = in_sizes[0] / 1024;
    const int nblocks = (N + NBLK - 1) / NBLK;   // 3125 for N=50000
    so2_linear_kernel<<<nblocks, NTHREADS, 0, stream>>>(x, wig, W0, b0, W1, W2, W3, out, N);
}